// IndependentPredictionHead_85521388798275
// MI455X (gfx1250) — compile-verified
//
#include <hip/hip_runtime.h>

// ---------------------------------------------------------------------------
// IndependentPredictionHead, fully fused, fp32 WMMA (V_WMMA_F32_16X16X4_F32)
//   hp[b,i,:]  = h[b,i,:] @ proj_w^T + proj_b          (GEMM 1, K=1024)
//   t [b,i,:]  = hp[b,i,:] @ lambda[b]                  (GEMM 2, K=1024)
//   diag[b,i]  = dot(t[b,i,:], u[i,:]) ; y = diag+bias, broadcast to 96
// One workgroup (8 wave32) owns 32 rows of one batch; hp stays in LDS.
// New this round:
//   * stage-0 h staging via GLOBAL_LOAD_ASYNC_TO_LDS_B128 (ASYNCcnt path)
//   * global_prefetch of lambda[b] into L2 overlapped with stage-1 compute
// ---------------------------------------------------------------------------

typedef __attribute__((ext_vector_type(2))) float v2f;
typedef __attribute__((ext_vector_type(8))) float v8f;

#define B_      64
#define D_      1024
#define K_      1024      // DMODEL
#define M_      1024
#define P_      96        // PRED_LEN
#define TI      32        // output rows per workgroup
#define LSTRIDE 1028      // padded LDS row stride (bank-conflict free, 16B aligned)

__device__ __forceinline__ v8f wmma_f32(v2f a, v2f b, v8f c) {
    // 8 args: (neg_a, A, neg_b, B, c_mod, C, reuse_a, reuse_b)
    return __builtin_amdgcn_wmma_f32_16x16x4_f32(
        false, a, false, b, (short)0, c, false, false);
}

// Async DMA: 16 bytes global -> LDS, tracked with ASYNCcnt (no VGPR transit).
__device__ __forceinline__ void async_copy_b128(const float* lds_ptr, const float* gptr) {
    const unsigned lds_off = (unsigned)(unsigned long long)lds_ptr;  // low 32b = LDS offset
    asm volatile("global_load_async_to_lds_b128 %0, %1, off"
                 :: "v"(lds_off), "v"(gptr)
                 : "memory");
}

__device__ __forceinline__ void wait_async_zero() {
    asm volatile("s_wait_asynccnt 0x0" ::: "memory");
}

extern "C" __global__ __launch_bounds__(256)
void iph_fused(const float* __restrict__ h,
               const float* __restrict__ lam,
               const float* __restrict__ u,
               const float* __restrict__ bias,
               const float* __restrict__ pw,
               const float* __restrict__ pb,
               float* __restrict__ out)
{
    extern __shared__ float smem[];
    float* hA = smem;                        // [TI][LSTRIDE]  staged h rows
    float* hp = smem + TI * LSTRIDE;         // [TI][LSTRIDE]  projected rows
    float* dg = smem + 2 * TI * LSTRIDE;     // [TI]           diag accumulator

    const int tid  = threadIdx.x;
    const int wave = tid >> 5;
    const int lane = tid & 31;
    const int half = lane >> 4;              // wave32: lanes 0-15 / 16-31
    const int l15  = lane & 15;

    const int b  = blockIdx.y;               // batch
    const int i0 = blockIdx.x * TI;          // first output row of this block

    const float* lamb = lam + (size_t)b * D_ * M_;

    // ---- stage 0: async DMA h[b, i0:i0+TI, :] into LDS (ASYNCcnt) ---------
    {
        const float* hgbase = h + ((size_t)b * D_ + i0) * K_;
        // TI*K_/4 = 8192 float4 chunks, 32 per thread
        for (int idx = tid; idx < TI * (K_ / 4); idx += 256) {
            const int r = idx / (K_ / 4);
            const int c = idx % (K_ / 4);
            async_copy_b128(&hA[r * LSTRIDE + c * 4],
                            hgbase + (size_t)r * K_ + c * 4);
        }
    }
    if (tid < TI) dg[tid] = 0.0f;

    // ---- warm L2 with lambda[b] while stage 1 runs (fire-and-forget) ------
    // 4 MB / 128B lines = 32768 lines; 128 per thread; no counter impact.
    {
        const char* lp = (const char*)lamb + (size_t)tid * 128;
        #pragma unroll 4
        for (int i = 0; i < 128; ++i)
            __builtin_prefetch(lp + (size_t)i * 256 * 128, 0, 0);   // global_prefetch_b8
    }

    wait_async_zero();          // LDS hA tile complete for this wave
    __syncthreads();            // ... and visible to all waves

    // ---- stage 1: hp = hA @ pw^T + pb  via 16x16x4 fp32 WMMA --------------
    // Each wave owns 8 of the 64 column tiles; 2 row-subtiles share the B frag.
    for (int nt = wave; nt < M_ / 16; nt += 8) {
        const int n = nt * 16 + l15;                 // proj_w row == hp column
        const float* bsrc = pw + (size_t)n * K_;
        v8f acc0 = {};
        v8f acc1 = {};
        #pragma unroll 8
        for (int k0 = 0; k0 < K_; k0 += 4) {
            const int ka = k0 + 2 * half;
            v2f bf = *(const v2f*)(bsrc + ka);                       // global (L2-resident)
            v2f a0 = *(const v2f*)&hA[(l15)      * LSTRIDE + ka];    // ds_load_b64
            v2f a1 = *(const v2f*)&hA[(16 + l15) * LSTRIDE + ka];
            acc0 = wmma_f32(a0, bf, acc0);
            acc1 = wmma_f32(a1, bf, acc1);
        }
        const float pbn = pb[n];
        #pragma unroll
        for (int p = 0; p < 8; ++p) {
            hp[(p + 8 * half)      * LSTRIDE + nt * 16 + l15] = acc0[p] + pbn;
            hp[(16 + p + 8 * half) * LSTRIDE + nt * 16 + l15] = acc1[p] + pbn;
        }
    }
    __syncthreads();

    // ---- stage 2: t = hp @ lambda[b]; fold immediately into diag ----------
    for (int mt = wave; mt < M_ / 16; mt += 8) {
        const int m = mt * 16 + l15;
        v8f acc0 = {};
        v8f acc1 = {};
        #pragma unroll 4
        for (int k0 = 0; k0 < D_; k0 += 4) {
            const int ka = k0 + 2 * half;
            v2f bf;
            bf.x = lamb[(size_t)(ka + 0) * M_ + m];   // coalesced 64B per 16 lanes
            bf.y = lamb[(size_t)(ka + 1) * M_ + m];
            v2f a0 = *(const v2f*)&hp[(l15)      * LSTRIDE + ka];
            v2f a1 = *(const v2f*)&hp[(16 + l15) * LSTRIDE + ka];
            acc0 = wmma_f32(a0, bf, acc0);
            acc1 = wmma_f32(a1, bf, acc1);
        }
        // diag[i] += sum_m t[i][m] * u[i][m] : butterfly-reduce the 16 columns
        #pragma unroll
        for (int p = 0; p < 8; ++p) {
            const int r0 = p + 8 * half;
            const int r1 = 16 + r0;
            float c0 = acc0[p] * u[(size_t)(i0 + r0) * M_ + m];
            float c1 = acc1[p] * u[(size_t)(i0 + r1) * M_ + m];
            #pragma unroll
            for (int s = 1; s < 16; s <<= 1) {
                c0 += __shfl_xor(c0, s);
                c1 += __shfl_xor(c1, s);
            }
            if (l15 == 0) {
                atomicAdd(&dg[r0], c0);               // ds_add_f32
                atomicAdd(&dg[r1], c1);
            }
        }
    }
    __syncthreads();

    // ---- stage 3: y[b,i,t] = diag[i] + bias[i], broadcast over PRED_LEN ---
    for (int idx = tid; idx < TI * P_; idx += 256) {
        const int r = idx / P_;
        const int t = idx % P_;
        out[((size_t)b * D_ + i0 + r) * P_ + t] = dg[r] + bias[i0 + r];
    }
}

extern "C" void kernel_launch(void* const* d_in, const int* in_sizes, int n_in,
                              void* d_out, int out_size, void* d_ws, size_t ws_size,
                              hipStream_t stream) {
    const float* h    = (const float*)d_in[0];   // [B, D, DMODEL]
    const float* lam  = (const float*)d_in[1];   // [B, D, M]
    const float* u    = (const float*)d_in[2];   // [D, M]
    const float* bias = (const float*)d_in[3];   // [D]
    const float* pw   = (const float*)d_in[4];   // [M, DMODEL]
    const float* pb   = (const float*)d_in[5];   // [M]
    float* out = (float*)d_out;                  // [B, D, PRED_LEN]

    dim3 grid(D_ / TI, B_);                      // 32 x 64 = 2048 workgroups
    dim3 block(256);                             // 8 wave32 per workgroup
    size_t smem = (size_t)(2 * TI * LSTRIDE + TI) * sizeof(float);  // ~257 KB of 320 KB WGP

    iph_fused<<<grid, block, smem, stream>>>(h, lam, u, bias, pw, pb, out);

    (void)in_sizes; (void)n_in; (void)out_size; (void)d_ws; (void)ws_size;
}